// Nms_38654705664161
// MI455X (gfx1250) — compile-verified
//
#include <hip/hip_runtime.h>
#include <stdint.h>

// ---------------------------------------------------------------------------
// Greedy batched NMS for MI455X (gfx1250, wave32).
//  - one 1024-thread workgroup (32 waves) per image (B=64 images)
//  - scores live in LDS (240KB < 320KB/WGP), suppressed in place
//  - boxes re-read from global each iteration (61MB total -> L2-resident)
//  - async-to-LDS (ASYNCcnt) for bulk score staging and the per-iteration
//    selected-box broadcast; global_prefetch keeps the box stream warm
//  - per iteration: one fused pass = suppress vs selected box + next argmax
// ---------------------------------------------------------------------------

#define BLOCK   1024
#define NPAD    61440                  // 15 * 4096, >= N=60000, multiple of 4*BLOCK
#define GROUPS  (NPAD / (BLOCK * 4))   // 15 float4-groups per thread
#define KDET    100
#define NEGV    (-1.0e10f)
#define NEGH    (-5.0e9f)              // NEG * 0.5 validity cutoff
#define SCORE_TH 0.05f

__device__ __forceinline__ unsigned lds_byte_off(const void* p) {
  // generic pointer to LDS: low 32 bits == LDS byte offset (LDS aperture)
  return (unsigned)(uintptr_t)p;
}

__global__ __launch_bounds__(BLOCK, 1) void nms_kernel(
    const float* __restrict__ scores,   // [B, N]
    const float* __restrict__ boxes,    // [B, N, 4] xyxy
    const int*   __restrict__ classes,  // [B, N]
    long long*   __restrict__ out_idx,  // [B, K]
    float*       __restrict__ out_sc,   // [B, K]
    float*       __restrict__ out_box,  // [B, K, 4]
    int*         __restrict__ out_cls,  // [B, K]
    int*         __restrict__ out_nv,   // [B]
    int N)
{
  __shared__ float ssc[NPAD];               // per-image score / suppression state
  __shared__ float wmax[32];
  __shared__ int   widx[32];
  __shared__ __align__(16) float bc[5];     // [0..3] selected box (async B128 target), [4] score
  __shared__ int   selsh;

  const int b    = blockIdx.x;
  const int tid  = threadIdx.x;
  const int lane = tid & 31;
  const int wave = tid >> 5;

  const float*  sc  = scores + (size_t)b * N;
  const float*  bx  = boxes  + (size_t)b * N * 4;
  const float4* bx4 = reinterpret_cast<const float4*>(bx);

  // ---- Phase 1: bulk async copy scores global -> LDS (16B per lane op) ----
  const int Na    = N & ~3;
  const int fullG = Na >> 12;                    // groups fully in range
  {
    unsigned loff = lds_byte_off(&ssc[tid << 2]);
    unsigned long long ga =
        (unsigned long long)(uintptr_t)(sc + (tid << 2));
    #pragma clang loop unroll(disable)
    for (int g = 0; g < fullG; ++g) {            // tight rolled issue loop
      asm volatile("global_load_async_to_lds_b128 %0, %1, off"
                   :: "v"(loff), "v"(ga) : "memory");
      loff += 4096u * 4u;                        // one group = 4096 floats
      ga   += 4096ull * 4ull;
    }
    for (int g = fullG; g < GROUPS; ++g) {       // boundary: per-lane guard
      int e = (tid << 2) + (g << 12);
      if (e < Na) {
        unsigned bl = lds_byte_off(&ssc[e]);
        unsigned long long bg = (unsigned long long)(uintptr_t)(sc + e);
        asm volatile("global_load_async_to_lds_b128 %0, %1, off"
                     :: "v"(bl), "v"(bg) : "memory");
      }
    }
  }
  asm volatile("s_wait_asynccnt 0" ::: "memory");
  __syncthreads();

  // ---- Phase 2: threshold in place + initial local argmax ----
  float lmax = NEGV;
  int   lidx = 0x7fffffff;
  for (int i = tid; i < NPAD; i += BLOCK) {
    float v;
    if (i < Na)     v = ssc[i];
    else if (i < N) v = sc[i];       // unaligned tail straight from global
    else            v = NEGV;        // padding
    v = (v > SCORE_TH) ? v : NEGV;
    ssc[i] = v;
    if (v > lmax) { lmax = v; lidx = i; }
  }

  int nv = 0;

  // ---- Phase 3: 100 greedy iterations (fused suppress + next-argmax) ----
  for (int t = 0; t < KDET; ++t) {
    // wave-level argmax (score desc, index asc on ties)
    float v = lmax; int ii = lidx;
    #pragma unroll
    for (int off = 16; off > 0; off >>= 1) {
      float ov = __shfl_xor(v, off, 32);
      int   oi = __shfl_xor(ii, off, 32);
      if (ov > v || (ov == v && oi < ii)) { v = ov; ii = oi; }
    }
    if (lane == 0) { wmax[wave] = v; widx[wave] = ii; }
    __syncthreads();

    if (wave == 0) {
      v = wmax[lane]; ii = widx[lane];
      #pragma unroll
      for (int off = 16; off > 0; off >>= 1) {
        float ov = __shfl_xor(v, off, 32);
        int   oi = __shfl_xor(ii, off, 32);
        if (ov > v || (ov == v && oi < ii)) { v = ov; ii = oi; }
      }
      if (lane == 0) {
        bc[4] = v;
        selsh = ii;
        if (v > NEGH) {
          // fetch selected box straight into the 16B-aligned LDS broadcast slot
          unsigned loff = lds_byte_off(&bc[0]);
          unsigned long long ga =
              (unsigned long long)(uintptr_t)(bx + (size_t)ii * 4);
          asm volatile("global_load_async_to_lds_b128 %0, %1, off\n\t"
                       "s_wait_asynccnt 0"
                       :: "v"(loff), "v"(ga) : "memory");
        } else {
          bc[0] = 0.f; bc[1] = 0.f; bc[2] = 0.f; bc[3] = 0.f;
        }
      }
    }
    __syncthreads();

    const int   sel   = selsh;
    const float svv   = bc[4];
    const bool  valid = svv > NEGH;
    const float x1 = bc[0], y1 = bc[1], x2 = bc[2], y2 = bc[3];

    if (tid == 0) {
      size_t o = (size_t)b * KDET + t;
      out_idx[o] = valid ? (long long)sel : -1LL;
      out_sc[o]  = valid ? svv : 0.0f;
      float4 ob;
      ob.x = valid ? x1 : 0.f;
      ob.y = valid ? y1 : 0.f;
      ob.z = valid ? x2 : 0.f;
      ob.w = valid ? y2 : 0.f;
      reinterpret_cast<float4*>(out_box)[o] = ob;
      out_cls[o] = valid ? classes[(size_t)b * N + sel] : -1;
      nv += valid ? 1 : 0;
    }

    // suppression pass fused with next iteration's local argmax
    // (compiler-chosen rolled form: keeps the hot loop I-cache resident)
    lmax = NEGV; lidx = 0x7fffffff;
    if (valid) {
      const float sarea = (x2 - x1) * (y2 - y1);
      for (int g = 0; g < GROUPS; ++g) {
        int e = (tid << 2) + (g << 12);
        // warm next group's box line (L2-resident stream)
        if (e + 4096 < N) __builtin_prefetch(&bx4[e + 4096], 0, 1);
        float4 s4 = *reinterpret_cast<float4*>(&ssc[e]);
        bool chg = false;
        #define NMS_DO(comp, kk)                                              \
          {                                                                   \
            float s = s4.comp;                                                \
            int gi = e + (kk);                                                \
            if (s > NEGH) { /* padding & suppressed are NEGV -> skipped */    \
              float4 bb = bx4[gi];                                            \
              float xx1 = fmaxf(x1, bb.x), yy1 = fmaxf(y1, bb.y);             \
              float xx2 = fminf(x2, bb.z), yy2 = fminf(y2, bb.w);             \
              float inter = fmaxf(xx2 - xx1, 0.f) * fmaxf(yy2 - yy1, 0.f);    \
              float area = (bb.z - bb.x) * (bb.w - bb.y);                     \
              float iou = inter / (sarea + area - inter + 1e-6f);             \
              if (iou > 0.5f || gi == sel) { s4.comp = NEGV; chg = true; }    \
              else if (s > lmax) { lmax = s; lidx = gi; }                     \
            }                                                                 \
          }
        NMS_DO(x, 0)
        NMS_DO(y, 1)
        NMS_DO(z, 2)
        NMS_DO(w, 3)
        #undef NMS_DO
        if (chg) *reinterpret_cast<float4*>(&ssc[e]) = s4;
      }
    }
    // no extra barrier needed: next iteration's two barriers order LDS reuse
  }

  if (tid == 0) out_nv[b] = nv;
}

extern "C" void kernel_launch(void* const* d_in, const int* in_sizes, int n_in,
                              void* d_out, int out_size, void* d_ws, size_t ws_size,
                              hipStream_t stream) {
  (void)n_in; (void)out_size; (void)d_ws; (void)ws_size;

  const float* scores  = (const float*)d_in[0];
  const float* boxes   = (const float*)d_in[1];
  const int*   classes = (const int*)d_in[2];

  const int B = 64;                 // reference setup: B=64 images
  const int N = in_sizes[0] / B;    // 60000
  const int K = KDET;

  // tuple outputs concatenated flat in return order
  char* base = (char*)d_out;
  long long* o_idx = (long long*)(base);
  size_t off = (size_t)B * K * sizeof(long long);
  float* o_sc = (float*)(base + off);
  off += (size_t)B * K * sizeof(float);
  float* o_box = (float*)(base + off);
  off += (size_t)B * K * 4 * sizeof(float);
  int* o_cls = (int*)(base + off);
  off += (size_t)B * K * sizeof(int);
  int* o_nv = (int*)(base + off);

  nms_kernel<<<dim3(B), dim3(BLOCK), 0, stream>>>(
      scores, boxes, classes, o_idx, o_sc, o_box, o_cls, o_nv, N);
}